// RNNTJoint_48172353192301
// MI455X (gfx1250) — compile-verified
//
#include <hip/hip_runtime.h>

// RNNT joint for MI455X (gfx1250), fp32 throughout via V_WMMA_F32_16X16X4_F32.
// Store-bound (268.7MB output): fuse bias+add+relu into the big GEMM, never
// materialize h, keep GEMM operands resident in L2, stream output with NT hint.
// Per-wave tile 32x64 (8 v8f accumulators) to stay spill-free.

typedef __attribute__((ext_vector_type(2))) float v2f;
typedef __attribute__((ext_vector_type(8))) float v8f;

constexpr int kBatch = 4;
constexpr int kT     = 256;
constexpr int kU     = 64;
constexpr int kHE    = 1024;
constexpr int kHJ    = 640;
constexpr int kV     = 1025;
constexpr int kVP    = 1088;   // 17*64, padded V for maskless B-frag loads

// workspace layout (float elements)
constexpr size_t kOffFproj = 0;                             // [1024][640]
constexpr size_t kOffGproj = (size_t)1024 * kHJ;            // [256][640]
constexpr size_t kOffWp    = kOffGproj + (size_t)256 * kHJ; // [320][2*kVP]

// ---------------------------------------------------------------------------
// Phase 0: pack W_out[640][1025] -> Wp[kk][2*n + j], j = k&1, kk = k>>1,
// zero-padded to kVP columns. B-frag = one aligned float2 load.
// ---------------------------------------------------------------------------
__global__ void pack_wout_kernel(const float* __restrict__ Wout,
                                 float* __restrict__ Wp) {
    int idx = blockIdx.x * 256 + threadIdx.x;
    if (idx >= kHJ * kVP) return;
    int k = idx / kVP;
    int n = idx % kVP;
    float v = (n < kV) ? Wout[(size_t)k * kV + n] : 0.0f;
    Wp[(size_t)(k >> 1) * (2 * kVP) + 2 * n + (k & 1)] = v;
}

// ---------------------------------------------------------------------------
// Projection GEMM: C[M,N] = A[M,K] (lda) @ Bw[K,N] (ldb) + bias[N].
// 32x64 tile per wave, 4 waves/block. M multiple of 128, N multiple of 64.
// ---------------------------------------------------------------------------
__global__ __launch_bounds__(128) void proj_gemm_kernel(
    const float* __restrict__ A, const float* __restrict__ Bw,
    const float* __restrict__ bias, float* __restrict__ C,
    int K, int lda, int ldb, int N) {
    const int lane = threadIdx.x & 31;
    const int wave = threadIdx.x >> 5;
    const int lm   = lane & 15;
    const int half = lane >> 4;
    const int row0 = (blockIdx.y * 4 + wave) * 32;
    const int col0 = blockIdx.x * 64;

    v8f acc[8];
#pragma unroll
    for (int nt = 0; nt < 4; ++nt) {
        float bz = bias[col0 + nt * 16 + lm];
        v8f c;
#pragma unroll
        for (int r = 0; r < 8; ++r) c[r] = bz;
        acc[nt]     = c;
        acc[4 + nt] = c;
    }

    const float* aptr = A + (size_t)(row0 + lm) * lda + 2 * half;
    const float* bptr = Bw + (size_t)(2 * half) * ldb + (col0 + lm);
    const int aoff = 16 * lda;   // second M-tile row offset (floats)

    for (int k = 0; k < K; k += 4) {
        v2f a0 = *(const v2f*)aptr;
        v2f a1 = *(const v2f*)(aptr + aoff);
        v2f bf[4];
#pragma unroll
        for (int nt = 0; nt < 4; ++nt) {
            v2f bv;
            bv[0] = bptr[nt * 16];
            bv[1] = bptr[nt * 16 + ldb];
            bf[nt] = bv;
        }
#pragma unroll
        for (int nt = 0; nt < 4; ++nt) {
            acc[nt] = __builtin_amdgcn_wmma_f32_16x16x4_f32(
                false, a0, false, bf[nt], (short)0, acc[nt], false, false);
            acc[4 + nt] = __builtin_amdgcn_wmma_f32_16x16x4_f32(
                false, a1, false, bf[nt], (short)0, acc[4 + nt], false, false);
        }
        aptr += 4;
        bptr += 4 * ldb;
    }

#pragma unroll
    for (int mt = 0; mt < 2; ++mt)
#pragma unroll
        for (int nt = 0; nt < 4; ++nt) {
            const int n = col0 + nt * 16 + lm;
#pragma unroll
            for (int r = 0; r < 8; ++r) {
                const int m = row0 + mt * 16 + 8 * half + r;
                C[(size_t)m * N + n] = acc[mt * 4 + nt][r];
            }
        }
}

// ---------------------------------------------------------------------------
// Fused joint GEMM: out[(bt*64+u), n] = relu(f_proj[bt] + g_proj[b*64+u]) @ W + b_out
// 32x64 tile per wave; all rows of a tile share one f_proj row (U==64).
// ---------------------------------------------------------------------------
__global__ __launch_bounds__(128) void rnnt_joint_kernel(
    const float* __restrict__ f_proj, const float* __restrict__ g_proj,
    const float* __restrict__ Wp, const float* __restrict__ b_out,
    float* __restrict__ out) {
    const int lane = threadIdx.x & 31;
    const int wave = threadIdx.x >> 5;
    const int lm   = lane & 15;
    const int half = lane >> 4;
    const int rb32 = blockIdx.y * 4 + wave;   // 32-row block, 0..2047
    const int bt   = rb32 >> 1;               // 0..1023
    const int usub = (rb32 & 1) * 32;         // u offset within the bt block
    const int b    = bt >> 8;                 // bt / T
    const int col0 = blockIdx.x * 64;

    const float* fptr = f_proj + (size_t)bt * kHJ + 2 * half;
    const float* gptr = g_proj + (size_t)(b * kU + usub + lm) * kHJ + 2 * half;
    const float* wptr = Wp + (size_t)half * (2 * kVP) + 2 * (col0 + lm);

    v8f acc[8];
#pragma unroll
    for (int nt = 0; nt < 4; ++nt) {
        const int n = col0 + nt * 16 + lm;
        const float bz = (n < kV) ? b_out[n] : 0.0f;
        v8f c;
#pragma unroll
        for (int r = 0; r < 8; ++r) c[r] = bz;
        acc[nt]     = c;
        acc[4 + nt] = c;
    }

    for (int k = 0; k < kHJ; k += 4) {
        const v2f fv = *(const v2f*)fptr;                 // shared bt row
        const v2f g0 = *(const v2f*)gptr;                 // u = usub+lm
        const v2f g1 = *(const v2f*)(gptr + 16 * kHJ);    // u = usub+16+lm
        v2f a0, a1;
        a0[0] = fv[0] + g0[0]; a0[1] = fv[1] + g0[1];
        a1[0] = fv[0] + g1[0]; a1[1] = fv[1] + g1[1];
        a0[0] = a0[0] > 0.0f ? a0[0] : 0.0f;              // fused ReLU
        a0[1] = a0[1] > 0.0f ? a0[1] : 0.0f;
        a1[0] = a1[0] > 0.0f ? a1[0] : 0.0f;
        a1[1] = a1[1] > 0.0f ? a1[1] : 0.0f;

        v2f bf[4];
#pragma unroll
        for (int nt = 0; nt < 4; ++nt)
            bf[nt] = *(const v2f*)(wptr + nt * 32);

#pragma unroll
        for (int nt = 0; nt < 4; ++nt) {
            acc[nt] = __builtin_amdgcn_wmma_f32_16x16x4_f32(
                false, a0, false, bf[nt], (short)0, acc[nt], false, false);
            acc[4 + nt] = __builtin_amdgcn_wmma_f32_16x16x4_f32(
                false, a1, false, bf[nt], (short)0, acc[4 + nt], false, false);
        }
        fptr += 4;
        gptr += 4;
        wptr += 4 * kVP;
    }

    // Streamed (non-temporal) output stores: keep L2 for resident operands.
#pragma unroll
    for (int nt = 0; nt < 4; ++nt) {
        const int n = col0 + nt * 16 + lm;
        if (n < kV) {
#pragma unroll
            for (int mt = 0; mt < 2; ++mt)
#pragma unroll
                for (int r = 0; r < 8; ++r) {
                    const int row = bt * 64 + usub + mt * 16 + 8 * half + r;
                    __builtin_nontemporal_store(
                        acc[mt * 4 + nt][r], &out[(size_t)row * kV + n]);
                }
        }
    }
}

// ---------------------------------------------------------------------------
extern "C" void kernel_launch(void* const* d_in, const int* in_sizes, int n_in,
                              void* d_out, int out_size, void* d_ws, size_t ws_size,
                              hipStream_t stream) {
    const float* f      = (const float*)d_in[0];
    const float* g      = (const float*)d_in[1];
    const float* W_enc  = (const float*)d_in[2];
    const float* b_enc  = (const float*)d_in[3];
    const float* W_pred = (const float*)d_in[4];
    const float* b_pred = (const float*)d_in[5];
    const float* W_out  = (const float*)d_in[6];
    const float* b_out  = (const float*)d_in[7];
    float* out = (float*)d_out;
    float* ws  = (float*)d_ws;

    float* f_proj = ws + kOffFproj;   // 1024 x 640
    float* g_proj = ws + kOffGproj;   // 256 x 640
    float* Wp     = ws + kOffWp;      // 320 x (2*1088)

    // Phase 0: pack W_out for maskless float2 B-frag loads.
    {
        const int total = kHJ * kVP;
        pack_wout_kernel<<<(total + 255) / 256, 256, 0, stream>>>(W_out, Wp);
    }
    // Phase 1: projections (tiny: ~1.5 GFLOP total, inputs ~5MB in L2).
    proj_gemm_kernel<<<dim3(kHJ / 64, 1024 / 128), 128, 0, stream>>>(
        f, W_enc, b_enc, f_proj, kHE, kHE, kHJ, kHJ);
    proj_gemm_kernel<<<dim3(kHJ / 64, 256 / 128), 128, 0, stream>>>(
        g, W_pred, b_pred, g_proj, kHJ, kHJ, kHJ, kHJ);
    // Phase 2: fused joint GEMM, 86 GFLOP, HBM-store bound (268.7MB out).
    rnnt_joint_kernel<<<dim3(17, 512), 128, 0, stream>>>(
        f_proj, g_proj, Wp, b_out, out);
}